// GCN_Message_14611478741197
// MI455X (gfx1250) — compile-verified
//
#include <hip/hip_runtime.h>
#include <hip/hip_bf16.h>

typedef __attribute__((ext_vector_type(2))) float v2f;
typedef __attribute__((ext_vector_type(8))) float v8f;

// ---------------------------------------------------------------- utilities

__device__ __forceinline__ void atom_add_f32(float* p, float v) {
  // relaxed agent-scope fp32 add -> native global_atomic_add_f32
  __hip_atomic_fetch_add(p, v, __ATOMIC_RELAXED, __HIP_MEMORY_SCOPE_AGENT);
}

__global__ void zero_kernel(float* __restrict__ p, long n) {
  long i = (long)blockIdx.x * blockDim.x + threadIdx.x;
  if (i < n) p[i] = 0.0f;
}

__global__ void degree_kernel(const int* __restrict__ dst, float* __restrict__ cnt, int E) {
  int e = blockIdx.x * blockDim.x + threadIdx.x;
  if (e < E) atom_add_f32(&cnt[dst[e]], 1.0f);
}

__global__ void invdeg_kernel(const float* __restrict__ cnt, float* __restrict__ inv, int n) {
  int i = blockIdx.x * blockDim.x + threadIdx.x;
  if (i < n) inv[i] = 1.0f / fmaxf(cnt[i], 1.0f);
}

// Edge gather + segment-sum via native fp32 atomics.
// One thread moves 4 features (float4 gather, 4 atomic adds).
// Feature tables (<=51.2MB) fit in the 192MB L2, so atomics stay on-chip.
template <int D>
__global__ void scatter_add_kernel(const float* __restrict__ x,
                                   const int* __restrict__ src,
                                   const int* __restrict__ dst,
                                   float* __restrict__ agg, int E) {
  constexpr int C = D / 4;
  long tid = (long)blockIdx.x * blockDim.x + threadIdx.x;
  long total = (long)E * C;
  if (tid >= total) return;
  int e = (int)(tid / C);
  int f = (int)(tid % C) * 4;
  int s = src[e];
  int d = dst[e];
  float4 v = *(const float4*)(x + (long)s * D + f);
  float* o = agg + (long)d * D + f;
  atom_add_f32(o + 0, v.x);
  atom_add_f32(o + 1, v.y);
  atom_add_f32(o + 2, v.z);
  atom_add_f32(o + 3, v.w);
}

__global__ void split_z2_kernel(const float* __restrict__ z2,
                                float* __restrict__ out01,
                                float* __restrict__ out2, int n) {
  int i = blockIdx.x * blockDim.x + threadIdx.x;
  if (i < n) {
    out01[2 * i + 0] = z2[3 * i + 0];
    out01[2 * i + 1] = z2[3 * i + 1];
    out2[i] = z2[3 * i + 2];
  }
}

// ------------------------------------------------- fused WMMA linear / SAGE
//
// HAS_AGG == true :  out = act( (agg*invdeg) @ Wl^T + b + x @ Wr^T )
// HAS_AGG == false:  out = act( x @ Wl^T + b )
//
// fp32 WMMA 16x16x4 (V_WMMA_F32_16X16X4_F32) preserves fp32 semantics.
// Block = 64 node rows (4 M-tiles); wave w owns output columns [16w,16w+16)
// and keeps 4 accumulators, reusing each B fragment across the 4 M-tiles
// (4x weight-traffic amortization + 8 WMMAs per 2 B-fragment loads).
// Weights are zero-padded in LDS up to DOUTP columns so the inner loop has
// NO predicated loads (keeps EXEC full, avoids saveexec-wrapped ds_loads).
// LDS row stride DIN+4: 16B-aligned rows (float4 staging) and conflict-free
// column-strided reads (bank = 4n + k mod 64 distinct over 16 lanes).
template <int DIN, int DOUT, int ACT, bool HAS_AGG>
__global__ __launch_bounds__(256) void wmma_linear_kernel(
    const float* __restrict__ agg, const float* __restrict__ invdeg,
    const float* __restrict__ x,
    const float* __restrict__ Wl, const float* __restrict__ bias,
    const float* __restrict__ Wr,
    float* __restrict__ out, float* __restrict__ out2, int nnodes) {
  constexpr int MT = 4;                     // M-tiles per block
  constexpr int ROWS = 16 * MT;             // node rows per block
  constexpr int DOUTP = (DOUT + 15) & ~15;  // padded output columns
  constexpr int LDW = DIN + 4;              // LDS row stride (floats)
  constexpr int C4 = DIN / 4;
  __shared__ float sWl[DOUTP * LDW];
  __shared__ float sWr[HAS_AGG ? DOUTP * LDW : 1];
  __shared__ float sA[ROWS * LDW];                // scaled mean (or x)
  __shared__ float sX[HAS_AGG ? ROWS * LDW : 1];  // root features

  const int tid = threadIdx.x;
  const int bs = blockDim.x;
  const int row0 = blockIdx.x * ROWS;

  const float4 f4z = make_float4(0.0f, 0.0f, 0.0f, 0.0f);

  // Stage weights (float4 global reads, float4 LDS writes, zero-pad rows)
  for (int i = tid; i < DOUTP * C4; i += bs) {
    int r = i / C4, c4 = (i - r * C4) * 4;
    float4 wl = (r < DOUT) ? *(const float4*)(Wl + (long)r * DIN + c4) : f4z;
    *(float4*)&sWl[r * LDW + c4] = wl;
    if constexpr (HAS_AGG) {
      float4 wr = (r < DOUT) ? *(const float4*)(Wr + (long)r * DIN + c4) : f4z;
      *(float4*)&sWr[r * LDW + c4] = wr;
    }
  }
  // Stage the ROWS-row A tile(s), applying 1/deg scaling for the mean term
  for (int i = tid; i < ROWS * C4; i += bs) {
    int r = i / C4, c4 = (i - r * C4) * 4;
    int m = row0 + r;
    if (m >= nnodes) m = nnodes - 1;
    if constexpr (HAS_AGG) {
      float sc = invdeg[m];
      float4 av = *(const float4*)(agg + (long)m * DIN + c4);
      av.x *= sc; av.y *= sc; av.z *= sc; av.w *= sc;
      *(float4*)&sA[r * LDW + c4] = av;
      *(float4*)&sX[r * LDW + c4] = *(const float4*)(x + (long)m * DIN + c4);
    } else {
      *(float4*)&sA[r * LDW + c4] = *(const float4*)(x + (long)m * DIN + c4);
    }
  }
  __syncthreads();

  const int wave = tid >> 5;
  const int lane = tid & 31;
  const int half = lane >> 4;  // K split: lanes 16-31 hold K+2/K+3
  const int l16 = lane & 15;
  const int n = wave * 16 + l16;  // n < DOUTP always (weights zero-padded)

  v8f acc[MT];
#pragma unroll
  for (int mt = 0; mt < MT; ++mt) acc[mt] = (v8f){};

#pragma unroll
  for (int k = 0; k < DIN; k += 4) {
    const int kk = k + 2 * half;
    v2f b;
    b.x = sWl[n * LDW + kk];
    b.y = sWl[n * LDW + kk + 1];
    v2f b2;
    if constexpr (HAS_AGG) {
      b2.x = sWr[n * LDW + kk];
      b2.y = sWr[n * LDW + kk + 1];
    }
#pragma unroll
    for (int mt = 0; mt < MT; ++mt) {
      const int r = mt * 16 + l16;
      v2f a;
      a.x = sA[r * LDW + kk];
      a.y = sA[r * LDW + kk + 1];
      acc[mt] = __builtin_amdgcn_wmma_f32_16x16x4_f32(false, a, false, b,
                                                      (short)0, acc[mt], false,
                                                      false);
      if constexpr (HAS_AGG) {
        v2f a2;
        a2.x = sX[r * LDW + kk];
        a2.y = sX[r * LDW + kk + 1];
        acc[mt] = __builtin_amdgcn_wmma_f32_16x16x4_f32(false, a2, false, b2,
                                                        (short)0, acc[mt],
                                                        false, false);
      }
    }
  }

  const bool colOK = (n < DOUT);
  const float bcol = colOK ? bias[colOK ? n : 0] : 0.0f;
#pragma unroll
  for (int mt = 0; mt < MT; ++mt) {
#pragma unroll
    for (int i = 0; i < 8; ++i) {
      int row = row0 + mt * 16 + i + 8 * half;  // C/D: VGPR i = rows i / i+8
      float v = acc[mt][i] + bcol;
      if (ACT == 1) v = tanhf(v);
      if (ACT == 2) v = (v > 0.0f) ? v : 0.01f * v;
      if (colOK && row < nnodes) {
        out[(long)row * DOUT + n] = v;
        if (out2) out2[(long)row * DOUT + n] = v;
      }
    }
  }
}

// ------------------------------------------------------------------- driver

static inline unsigned cdiv(long a, long b) { return (unsigned)((a + b - 1) / b); }

extern "C" void kernel_launch(void* const* d_in, const int* in_sizes, int n_in,
                              void* d_out, int out_size, void* d_ws, size_t ws_size,
                              hipStream_t stream) {
  const float* x = (const float*)d_in[0];
  const int* ei = (const int*)d_in[1];
  const int N = in_sizes[0] / 64;
  const int E = in_sizes[1] / 2;
  const int* src = ei;
  const int* dst = ei + E;

  const float* g1Wl = (const float*)d_in[2];
  const float* g1b  = (const float*)d_in[3];
  const float* g1Wr = (const float*)d_in[4];
  const float* g2Wl = (const float*)d_in[5];
  const float* g2b  = (const float*)d_in[6];
  const float* g2Wr = (const float*)d_in[7];
  const float* g3Wl = (const float*)d_in[8];
  const float* g3b  = (const float*)d_in[9];
  const float* g3Wr = (const float*)d_in[10];
  const float* trW  = (const float*)d_in[11];
  const float* trb  = (const float*)d_in[12];
  const float* d1W  = (const float*)d_in[13];
  const float* d1b  = (const float*)d_in[14];
  const float* d2Wl = (const float*)d_in[15];
  const float* d2b  = (const float*)d_in[16];
  const float* d2Wr = (const float*)d_in[17];
  const float* d3Wl = (const float*)d_in[18];
  const float* d3b  = (const float*)d_in[19];
  const float* d3Wr = (const float*)d_in[20];
  const float* d4Wl = (const float*)d_in[21];
  const float* d4b  = (const float*)d_in[22];
  const float* d4Wr = (const float*)d_in[23];

  // workspace (floats): cnt[N] inv[N] bufA[N*128] bufB[N*128] bufC[N*128] z2[N*3]
  float* ws   = (float*)d_ws;
  float* cnt  = ws;
  float* inv  = ws + N;
  float* bufA = ws + 2 * (size_t)N;
  float* bufB = bufA + (size_t)N * 128;
  float* bufC = bufB + (size_t)N * 128;
  float* z2   = bufC + (size_t)N * 128;

  // output layout: z2[:, :2] (2N) | z2[:,2] (N) | mu (64N) | logvar (64N)
  float* outZ01 = (float*)d_out;
  float* outZ2  = outZ01 + 2 * (size_t)N;
  float* outMu  = outZ2 + N;
  float* outLv  = outMu + (size_t)N * 64;

  const unsigned gN64 = (unsigned)((N + 63) / 64);  // 64 rows per block

  // degrees (shared by every SAGE layer)
  zero_kernel<<<cdiv(N, 256), 256, 0, stream>>>(cnt, N);
  degree_kernel<<<cdiv(E, 256), 256, 0, stream>>>(dst, cnt, E);
  invdeg_kernel<<<cdiv(N, 256), 256, 0, stream>>>(cnt, inv, N);

  // encoder L1: SAGE 64 -> 128, tanh
  zero_kernel<<<cdiv((long)N * 64, 256), 256, 0, stream>>>(bufA, (long)N * 64);
  scatter_add_kernel<64><<<cdiv((long)E * 16, 256), 256, 0, stream>>>(x, src, dst, bufA, E);
  wmma_linear_kernel<64, 128, 1, true><<<gN64, 256, 0, stream>>>(
      bufA, inv, x, g1Wl, g1b, g1Wr, bufB, nullptr, N);

  // encoder L2: SAGE 128 -> 128, tanh
  zero_kernel<<<cdiv((long)N * 128, 256), 256, 0, stream>>>(bufA, (long)N * 128);
  scatter_add_kernel<128><<<cdiv((long)E * 32, 256), 256, 0, stream>>>(bufB, src, dst, bufA, E);
  wmma_linear_kernel<128, 128, 1, true><<<gN64, 256, 0, stream>>>(
      bufA, inv, bufB, g2Wl, g2b, g2Wr, bufC, nullptr, N);

  // encoder L3: SAGE 128 -> 128, tanh
  zero_kernel<<<cdiv((long)N * 128, 256), 256, 0, stream>>>(bufA, (long)N * 128);
  scatter_add_kernel<128><<<cdiv((long)E * 32, 256), 256, 0, stream>>>(bufC, src, dst, bufA, E);
  wmma_linear_kernel<128, 128, 1, true><<<gN64, 256, 0, stream>>>(
      bufA, inv, bufC, g3Wl, g3b, g3Wr, bufB, nullptr, N);

  // mu = h3 @ trW^T + trb ; logvar = mu (written to both output regions)
  wmma_linear_kernel<128, 64, 0, false><<<gN64, 128, 0, stream>>>(
      nullptr, nullptr, bufB, trW, trb, nullptr, outMu, outLv, N);

  // decoder L1: dense 64 -> 128, leaky
  wmma_linear_kernel<64, 128, 2, false><<<gN64, 256, 0, stream>>>(
      nullptr, nullptr, outMu, d1W, d1b, nullptr, bufA, nullptr, N);

  // decoder L2: SAGE 128 -> 128, leaky
  zero_kernel<<<cdiv((long)N * 128, 256), 256, 0, stream>>>(bufC, (long)N * 128);
  scatter_add_kernel<128><<<cdiv((long)E * 32, 256), 256, 0, stream>>>(bufA, src, dst, bufC, E);
  wmma_linear_kernel<128, 128, 2, true><<<gN64, 256, 0, stream>>>(
      bufC, inv, bufA, d2Wl, d2b, d2Wr, bufB, nullptr, N);

  // decoder L3: SAGE 128 -> 64, tanh
  zero_kernel<<<cdiv((long)N * 128, 256), 256, 0, stream>>>(bufC, (long)N * 128);
  scatter_add_kernel<128><<<cdiv((long)E * 32, 256), 256, 0, stream>>>(bufB, src, dst, bufC, E);
  wmma_linear_kernel<128, 64, 1, true><<<gN64, 128, 0, stream>>>(
      bufC, inv, bufB, d3Wl, d3b, d3Wr, bufA, nullptr, N);

  // decoder L4: SAGE 64 -> 3, linear
  zero_kernel<<<cdiv((long)N * 64, 256), 256, 0, stream>>>(bufC, (long)N * 64);
  scatter_add_kernel<64><<<cdiv((long)E * 16, 256), 256, 0, stream>>>(bufA, src, dst, bufC, E);
  wmma_linear_kernel<64, 3, 0, true><<<gN64, 32, 0, stream>>>(
      bufC, inv, bufA, d4Wl, d4b, d4Wr, z2, nullptr, N);

  // scatter z2 into the two output slices
  split_z2_kernel<<<cdiv(N, 256), 256, 0, stream>>>(z2, outZ01, outZ2, N);
}